// Sam3MaskPredictor_24163486007494
// MI455X (gfx1250) — compile-verified
//
#include <hip/hip_runtime.h>

typedef float v2f __attribute__((ext_vector_type(2)));
typedef float v4f __attribute__((ext_vector_type(4)));
typedef float v8f __attribute__((ext_vector_type(8)));

#define HIDDEN 256
#define NQ 200
#define NQ_PAD 208       // 13 tiles of 16
#define HW 65536
#define MT 13            // M tiles
#define KC 32            // K chunk staged in LDS per buffer
#define LDS_STRIDE (KC + 4)  // 36 dwords: bank-conflict-free b128 reads

#if __has_builtin(__builtin_amdgcn_global_load_async_to_lds_b128)
#define ASYNC_STAGE 1
// Builtin signature (from hipcc diagnostic): param0 = AS(1) v4i*, then LDS
// pointer, imm offset, cpol.
typedef int v4i_vs __attribute__((vector_size(16)));
typedef __attribute__((address_space(1))) v4i_vs* gv4_t;
typedef __attribute__((address_space(3))) v4i_vs* lv4_t;
#else
#define ASYNC_STAGE 0
#endif

#if ASYNC_STAGE
#if __has_builtin(__builtin_amdgcn_s_wait_asynccnt)
#define WAIT_ASYNC() __builtin_amdgcn_s_wait_asynccnt(0)
#else
#define WAIT_ASYNC() asm volatile("s_wait_asynccnt 0x0" ::: "memory")
#endif
#else
#define WAIT_ASYNC()
#endif

static __device__ __forceinline__ v8f wmma_f32(v2f a, v2f b, v8f c) {
  // (neg_a, A, neg_b, B, c_mod, C, reuse_a, reuse_b)
  return __builtin_amdgcn_wmma_f32_16x16x4_f32(false, a, false, b, (short)0, c,
                                               false, false);
}

// ---------------------------------------------------------------------------
// Kernel 1: mask_embed = (relu(queries @ w1^T + b1)) @ w2^T + b2
// grid (13, 4), block 256 (8 waves). Rows >= 200 stored as zeros (padding).
// ---------------------------------------------------------------------------
__global__ __launch_bounds__(256) void sam3_mlp_kernel(
    const float* __restrict__ queries, const float* __restrict__ w1,
    const float* __restrict__ b1v, const float* __restrict__ w2,
    const float* __restrict__ b2v, float* __restrict__ embed) {
  __shared__ float hs[16][HIDDEN + 8];

  const int lane = threadIdx.x & 31;
  const int wave = threadIdx.x >> 5;
  const int col  = lane & 15;
  const int half = lane >> 4;
  const int m0   = blockIdx.x * 16;
  const int b    = blockIdx.y;

  const int qrow = m0 + col;
  const int qc   = qrow < NQ ? qrow : (NQ - 1);
  const float* aptr = queries + ((size_t)b * NQ + qc) * HIDDEN + 2 * half;

  for (int t = 0; t < 2; ++t) {
    const int n0 = (wave * 2 + t) * 16;
    const float* bptr = w1 + (size_t)(n0 + col) * HIDDEN + 2 * half;
    v8f acc = {};
#pragma unroll 8
    for (int k0 = 0; k0 < HIDDEN; k0 += 4) {
      v2f a  = *(const v2f*)(aptr + k0);
      v2f bb = *(const v2f*)(bptr + k0);
      acc = wmma_f32(a, bb, acc);
    }
    const float bias = b1v[n0 + col];
#pragma unroll
    for (int i = 0; i < 8; ++i) {
      float v = acc[i] + bias;
      hs[8 * half + i][n0 + col] = v > 0.f ? v : 0.f;
    }
  }
  __syncthreads();

  for (int t = 0; t < 2; ++t) {
    const int n0 = (wave * 2 + t) * 16;
    const float* bptr = w2 + (size_t)(n0 + col) * HIDDEN + 2 * half;
    v8f acc = {};
#pragma unroll 8
    for (int k0 = 0; k0 < HIDDEN; k0 += 4) {
      v2f a  = *(const v2f*)(&hs[col][k0 + 2 * half]);
      v2f bb = *(const v2f*)(bptr + k0);
      acc = wmma_f32(a, bb, acc);
    }
    const float bias = b2v[n0 + col];
#pragma unroll
    for (int i = 0; i < 8; ++i) {
      const int q = m0 + 8 * half + i;
      const float v = acc[i] + bias;
      embed[((size_t)b * NQ_PAD + q) * HIDDEN + (n0 + col)] =
          (q < NQ) ? v : 0.f;
    }
  }
}

// ---------------------------------------------------------------------------
// Stage one KC-wide K-chunk of the 208x256 embed tile into LDS.
// 208*KC/4 = 1664 b128 transfers across 256 threads, via async-to-LDS.
// ---------------------------------------------------------------------------
static __device__ __forceinline__ void stage_chunk(
    const float* __restrict__ gsrc,  // embed[b] + k0 (row stride HIDDEN)
    float* lbase,                    // &As[buf][0][0] (generic ptr to LDS)
    int tid) {
  constexpr int TOTAL = NQ_PAD * (KC / 4);
  for (int t = tid; t < TOTAL; t += 256) {
    const int row  = t >> 3;          // KC/4 == 8 transfers per row
    const int koff = (t & 7) * 4;
    const float* g = gsrc + (size_t)row * HIDDEN + koff;
    float* l = lbase + row * LDS_STRIDE + koff;
#if ASYNC_STAGE
    __builtin_amdgcn_global_load_async_to_lds_b128((gv4_t)g, (lv4_t)l, 0, 0);
#else
    *(v4f*)l = *(const v4f*)g;
#endif
  }
}

// ---------------------------------------------------------------------------
// Kernel 2: pred[b,q,hw] = sum_c embed[b,q,c] * mf[b,c,hw]
// grid (512, 4), block 256 (8 waves). Wave owns 16 hw-columns and all 13
// q-tiles; mask_features streamed from HBM exactly once (non-temporal).
// A staged in LDS, double-buffered via async global->LDS copies.
// ---------------------------------------------------------------------------
__global__ __launch_bounds__(256) void sam3_mask_kernel(
    const float* __restrict__ embed, const float* __restrict__ mf,
    float* __restrict__ out) {
  __shared__ __align__(16) float As[2][NQ_PAD][LDS_STRIDE];  // ~60 KB

  const int tid  = threadIdx.x;
  const int lane = tid & 31;
  const int wave = tid >> 5;
  const int col  = lane & 15;
  const int half = lane >> 4;
  const int b    = blockIdx.y;
  const int n    = blockIdx.x * 128 + wave * 16 + col;

  const float* bbase = mf + (size_t)b * HIDDEN * HW + n;   // + k*HW per row
  const float* abase = embed + (size_t)b * NQ_PAD * HIDDEN;

  v8f acc[MT];
#pragma unroll
  for (int m = 0; m < MT; ++m) acc[m] = (v8f){};

  stage_chunk(abase, &As[0][0][0], tid);  // prefetch chunk 0

  int cur = 0;
  for (int c = 0; c < HIDDEN / KC; ++c) {
    WAIT_ASYNC();
    __syncthreads();  // chunk c fully resident in As[cur]

    if (c + 1 < HIDDEN / KC)
      stage_chunk(abase + (c + 1) * KC, &As[cur ^ 1][0][0], tid);

#pragma unroll
    for (int s = 0; s < KC / 8; ++s) {
      const int k8 = c * KC + s * 8;
      // B frags: rows {k8+4h, +1} (even step), {k8+4h+2, +3} (odd step).
      const float* bp = bbase + (size_t)(k8 + 4 * half) * HW;
      v2f be, bo;
      be.x = __builtin_nontemporal_load(bp + 0 * (size_t)HW);
      be.y = __builtin_nontemporal_load(bp + 1 * (size_t)HW);
      bo.x = __builtin_nontemporal_load(bp + 2 * (size_t)HW);
      bo.y = __builtin_nontemporal_load(bp + 3 * (size_t)HW);
#pragma unroll
      for (int m = 0; m < MT; ++m) {
        // A from LDS: b128 -> K slots {.., +3}; xy even step, zw odd step.
        const v4f a4 =
            *(const v4f*)&As[cur][m * 16 + col][s * 8 + 4 * half];
        v2f ae, ao;
        ae.x = a4.x; ae.y = a4.y;
        ao.x = a4.z; ao.y = a4.w;
        acc[m] = wmma_f32(ae, be, acc[m]);
        acc[m] = wmma_f32(ao, bo, acc[m]);
      }
    }
    __syncthreads();  // all waves done with As[cur] before it is restaged
    cur ^= 1;
  }

  // Epilogue: C element i -> row m*16 + 8*half + i, col n. NT stores.
  float* obase = out + (size_t)b * NQ * HW + n;
#pragma unroll
  for (int m = 0; m < MT; ++m) {
#pragma unroll
    for (int i = 0; i < 8; ++i) {
      const int q = m * 16 + 8 * half + i;
      if (q < NQ)
        __builtin_nontemporal_store(acc[m][i], obase + (size_t)q * HW);
    }
  }
}

extern "C" void kernel_launch(void* const* d_in, const int* in_sizes, int n_in,
                              void* d_out, int out_size, void* d_ws,
                              size_t ws_size, hipStream_t stream) {
  const float* queries = (const float*)d_in[0];  // (4, 200, 256)
  const float* mf      = (const float*)d_in[1];  // (4, 256, 256, 256)
  const float* w1      = (const float*)d_in[2];  // (256, 256)
  const float* b1v     = (const float*)d_in[3];  // (256,)
  const float* w2      = (const float*)d_in[4];  // (256, 256)
  const float* b2v     = (const float*)d_in[5];  // (256,)
  float* out   = (float*)d_out;                  // (4, 200, 256, 256)
  float* embed = (float*)d_ws;                   // 4*208*256 f32 = 832 KB

  sam3_mlp_kernel<<<dim3(MT, 4), 256, 0, stream>>>(queries, w1, b1v, w2, b2v,
                                                   embed);
  sam3_mask_kernel<<<dim3(HW / 128, 4), 256, 0, stream>>>(embed, mf, out);
}